// SelfModifyingLayer_45071386804408
// MI455X (gfx1250) — compile-verified
//
#include <hip/hip_runtime.h>
#include <hip/hip_bf16.h>

// Problem dims (fixed by the reference)
#define BATCH 16
#define SEQ   4096
#define DDIM  512
#define HDIM  256
#define CHUNK 64
#define NCH   (SEQ / CHUNK)      // 64 chunks
#define USTR  66                 // padded LDS stride for 16x64 tiles (even, conflict-free)

typedef __attribute__((ext_vector_type(2))) float v2f;
typedef __attribute__((ext_vector_type(8))) float v8f;

static __device__ __forceinline__ v8f wmma_f32(v2f a, v2f b, v8f c) {
    // D = A(16x4) * B(4x16) + C, fp32, wave32
    return __builtin_amdgcn_wmma_f32_16x16x4_f32(false, a, false, b, (short)0, c, false, false);
}

// ---------------------------------------------------------------------------
// Kernel 1: fused dual-gate MLP via fp32 WMMA, software-pipelined loads.
// ---------------------------------------------------------------------------
__global__ __launch_bounds__(256) void gate_kernel(
    const float* __restrict__ X,
    const float* __restrict__ w1a, const float* __restrict__ b1a,
    const float* __restrict__ w2a, const float* __restrict__ b2a,
    const float* __restrict__ w1e, const float* __restrict__ b1e,
    const float* __restrict__ w2e, const float* __restrict__ b2e,
    float* __restrict__ alpha, float* __restrict__ eta)
{
    const int lane = threadIdx.x & 31;
    const int wave = threadIdx.x >> 5;
    const int m0   = (blockIdx.x * 8 + wave) * 16;
    const int half = lane >> 4;
    const int lid  = lane & 15;
    const int koff = half ? 2 : 0;

    const float* xr = X + (size_t)(m0 + lid) * DDIM + koff;

    float rowA[8], rowE[8];
    #pragma unroll
    for (int v = 0; v < 8; ++v) { rowA[v] = 0.f; rowE[v] = 0.f; }

    for (int ng = 0; ng < 4; ++ng) {
        v8f accA[4], accE[4];
        const float* wAr[4];
        const float* wEr[4];
        #pragma unroll
        for (int t = 0; t < 4; ++t) {
            accA[t] = (v8f){0.f,0.f,0.f,0.f,0.f,0.f,0.f,0.f};
            accE[t] = (v8f){0.f,0.f,0.f,0.f,0.f,0.f,0.f,0.f};
            const int n0 = (ng * 4 + t) * 16;
            wAr[t] = w1a + (size_t)(n0 + lid) * DDIM + koff;
            wEr[t] = w1e + (size_t)(n0 + lid) * DDIM + koff;
        }
        // depth-1 software pipeline: prefetch k+4 while WMMAs consume k.
        // unroll 2 so the cur/next buffer rotation vanishes via renaming.
        v2f a = *(const v2f*)xr;
        v2f bA[4], bE[4];
        #pragma unroll
        for (int t = 0; t < 4; ++t) { bA[t] = *(const v2f*)wAr[t]; bE[t] = *(const v2f*)wEr[t]; }
        #pragma unroll 2
        for (int k0 = 0; k0 < DDIM; k0 += 4) {
            const int kn = (k0 + 4) & (DDIM - 1);
            v2f a_n = *(const v2f*)(xr + kn);
            v2f bA_n[4], bE_n[4];
            #pragma unroll
            for (int t = 0; t < 4; ++t) {
                bA_n[t] = *(const v2f*)(wAr[t] + kn);
                bE_n[t] = *(const v2f*)(wEr[t] + kn);
            }
            #pragma unroll
            for (int t = 0; t < 4; ++t) {
                accA[t] = wmma_f32(a, bA[t], accA[t]);
                accE[t] = wmma_f32(a, bE[t], accE[t]);
            }
            a = a_n;
            #pragma unroll
            for (int t = 0; t < 4; ++t) { bA[t] = bA_n[t]; bE[t] = bE_n[t]; }
        }
        #pragma unroll
        for (int t = 0; t < 4; ++t) {
            const int n = (ng * 4 + t) * 16 + lid;
            const float c1A = b1a[n], c2A = w2a[n];
            const float c1E = b1e[n], c2E = w2e[n];
            #pragma unroll
            for (int v = 0; v < 8; ++v) {
                float hA = accA[t][v] + c1A; hA = hA > 0.f ? hA : 0.f;
                rowA[v] = __builtin_fmaf(hA, c2A, rowA[v]);
                float hE = accE[t][v] + c1E; hE = hE > 0.f ? hE : 0.f;
                rowE[v] = __builtin_fmaf(hE, c2E, rowE[v]);
            }
        }
    }
    #pragma unroll
    for (int v = 0; v < 8; ++v) {
        #pragma unroll
        for (int m = 1; m < 16; m <<= 1) {
            rowA[v] += __shfl_xor(rowA[v], m, 32);
            rowE[v] += __shfl_xor(rowE[v], m, 32);
        }
    }
    if (lid == 0) {
        const float bb2a = b2a[0], bb2e = b2e[0];
        const int rbase = m0 + half * 8;
        #pragma unroll
        for (int v = 0; v < 8; ++v) {
            alpha[rbase + v] = 1.f / (1.f + __expf(-(rowA[v] + bb2a)));
            eta[rbase + v]   = 1.f / (1.f + __expf(-(rowE[v] + bb2e)));
        }
    }
}

// ---------------------------------------------------------------------------
// Kernel 2: per-(batch,chunk) preprocessing. Gram S = X X^T via WMMA, prefix
// products, L = diag-ratio * e * S, unit-triangular inverse T = (I+L)^-1
// (64 independent column solves), P^T = (diag(A_{i-1}) T)^T, q, A_fin.
// All 1024 (b,chunk) blocks run in parallel (independent of M).
// ---------------------------------------------------------------------------
__global__ __launch_bounds__(256) void prep_kernel(
    const float* __restrict__ X,
    const float* __restrict__ alpha, const float* __restrict__ eta,
    float* __restrict__ PT, float* __restrict__ qv, float* __restrict__ afinv)
{
    __shared__ float Sm[CHUNK * CHUNK];
    __shared__ float Tm[CHUNK * CHUNK];
    __shared__ float ap[CHUNK];   // A_{i-1} (exclusive prefix product)
    __shared__ float Af[CHUNK];   // A_i (inclusive)
    __shared__ float ev[CHUNK];
    __shared__ float s_afin;

    const int tid  = threadIdx.x;
    const int lane = tid & 31;
    const int wave = tid >> 5;
    const int half = lane >> 4;
    const int lid  = lane & 15;
    const int koff = half ? 2 : 0;

    const int bc = blockIdx.x;               // b*NCH + c
    const float* xc = X + (size_t)bc * CHUNK * DDIM;   // contiguous [B][S][D]
    const float* ag = alpha + (size_t)bc * CHUNK;
    const float* eg = eta   + (size_t)bc * CHUNK;

    if (tid == 0) {
        float run = 1.f;
        for (int i = 0; i < CHUNK; ++i) {
            ap[i] = run;
            run *= ag[i];
            Af[i] = run;
            ev[i] = eg[i];
        }
        s_afin = run;
    }

    // S = X X^T : 16 tiles of 16x16, 2 per wave, K=512 via WMMA
    #pragma unroll
    for (int r = 0; r < 2; ++r) {
        const int tt = wave + 8 * r;
        const int jt = tt >> 2, it = tt & 3;
        const float* ar = xc + (size_t)(jt * 16 + lid) * DDIM + koff;
        const float* br = xc + (size_t)(it * 16 + lid) * DDIM + koff;
        v8f acc = (v8f){0.f,0.f,0.f,0.f,0.f,0.f,0.f,0.f};
        v2f a = *(const v2f*)ar;
        v2f bb = *(const v2f*)br;
        #pragma unroll 2
        for (int k0 = 0; k0 < DDIM; k0 += 4) {
            const int kn = (k0 + 4) & (DDIM - 1);
            v2f an = *(const v2f*)(ar + kn);
            v2f bn = *(const v2f*)(br + kn);
            acc = wmma_f32(a, bb, acc);
            a = an; bb = bn;
        }
        #pragma unroll
        for (int v = 0; v < 8; ++v)
            Sm[(jt * 16 + v + 8 * half) * CHUNK + it * 16 + lid] = acc[v];
    }
    __syncthreads();

    // L[j][i] = (A_{i-1}/A_j) * e_j * S[j][i]  (strictly upper), in place.
    for (int t = tid; t < CHUNK * CHUNK; t += 256) {
        const int j = t >> 6, i = t & 63;
        float val = 0.f;
        if (j < i) val = (ap[i] / Af[j]) * ev[j] * Sm[t];
        Sm[t] = val;
    }
    __syncthreads();

    // T = (I+L)^-1, unit upper triangular; thread i solves column i.
    if (tid < CHUNK) {
        const int i = tid;
        Tm[i * CHUNK + i] = 1.f;
        for (int j = i - 1; j >= 0; --j) {
            float s = 0.f;
            for (int k = j + 1; k <= i; ++k)
                s = __builtin_fmaf(Sm[j * CHUNK + k], Tm[k * CHUNK + i], s);
            Tm[j * CHUNK + i] = -s;
        }
    }
    __syncthreads();

    // PT[i][j] = A_{j-1} * T[j][i]  (transposed so GEMM B-fragments are contiguous)
    float* ptc = PT + (size_t)bc * (CHUNK * CHUNK);
    for (int t = tid; t < CHUNK * CHUNK; t += 256) {
        const int i = t >> 6, j = t & 63;
        ptc[t] = (j <= i) ? ap[j] * Tm[j * CHUNK + i] : 0.f;
    }
    if (tid < CHUNK) qv[(size_t)bc * CHUNK + tid] = (s_afin / Af[tid]) * ev[tid];
    if (tid == 0)    afinv[bc] = s_afin;
}

// ---------------------------------------------------------------------------
// Kernel 3: chunked fast-weight scan, entirely on WMMA. Each WG owns a 16-row
// tile of M, resident in LDS for all 64 chunks (M never touches global).
// Per chunk: U = M X^T (K=512) -> Y = U P (K=64) -> out = Y + x ->
// M = A_fin*M - (Y diag(q)) X (K=64). Two waves split the i-tiles / col-tiles.
// ---------------------------------------------------------------------------
__global__ __launch_bounds__(64) void scan_kernel(
    const float* __restrict__ X, const float* __restrict__ Mem,
    const float* __restrict__ PT, const float* __restrict__ qv,
    const float* __restrict__ afinv, float* __restrict__ out)
{
    __shared__ float Mlds[16 * DDIM];     // 32 KB: this WG's 16 rows of M
    __shared__ float Ubuf[16 * USTR];     // U tile, then Y tile (layout shuttle)

    const int tid  = threadIdx.x;
    const int lane = tid & 31;
    const int wave = tid >> 5;            // 0..1
    const int half = lane >> 4;
    const int lid  = lane & 15;
    const int koff = half ? 2 : 0;

    const int b    = blockIdx.x >> 5;
    const int rt   = blockIdx.x & 31;
    const int row0 = rt * 16;

    { // init M tile from the broadcast initial memory (rows are contiguous)
        const float4* src = (const float4*)(Mem + (size_t)row0 * DDIM);
        float4* dst = (float4*)Mlds;
        for (int i = tid; i < 16 * DDIM / 4; i += 64) dst[i] = src[i];
    }
    __syncthreads();

    const float* xb = X   + (size_t)b * SEQ * DDIM;
    float*       ob = out + (size_t)b * SEQ * DDIM;

    for (int c = 0; c < NCH; ++c) {
        const float* xc  = xb + (size_t)c * CHUNK * DDIM;
        const float* ptc = PT + (size_t)(b * NCH + c) * (CHUNK * CHUNK);
        const float* qc  = qv + (size_t)(b * NCH + c) * CHUNK;
        const float  af  = afinv[b * NCH + c];

        // ---- U = M X^T : this wave's two 16x16 i-tiles, K = 512 ----
        v8f acc0 = (v8f){0.f,0.f,0.f,0.f,0.f,0.f,0.f,0.f};
        v8f acc1 = acc0;
        {
            const float* mr  = Mlds + lid * DDIM + koff;                    // A from LDS
            const float* xr0 = xc + (size_t)(wave * 32 + lid) * DDIM + koff;
            const float* xr1 = xc + (size_t)(wave * 32 + 16 + lid) * DDIM + koff;
            v2f aM = *(const v2f*)mr;
            v2f b0 = *(const v2f*)xr0;
            v2f b1 = *(const v2f*)xr1;
            #pragma unroll 2
            for (int k0 = 0; k0 < DDIM; k0 += 4) {
                const int kn = (k0 + 4) & (DDIM - 1);
                v2f aMn = *(const v2f*)(mr + kn);
                v2f b0n = *(const v2f*)(xr0 + kn);
                v2f b1n = *(const v2f*)(xr1 + kn);
                acc0 = wmma_f32(aM, b0, acc0);
                acc1 = wmma_f32(aM, b1, acc1);
                aM = aMn; b0 = b0n; b1 = b1n;
            }
        }
        #pragma unroll
        for (int v = 0; v < 8; ++v) {
            Ubuf[(v + 8 * half) * USTR + wave * 32 + lid]      = acc0[v];
            Ubuf[(v + 8 * half) * USTR + wave * 32 + 16 + lid] = acc1[v];
        }
        __syncthreads();

        // ---- Y = U * P : K = 64 (A from LDS shuttle, B = PT from global) ----
        v8f y0 = (v8f){0.f,0.f,0.f,0.f,0.f,0.f,0.f,0.f};
        v8f y1 = y0;
        {
            const float* ur = Ubuf + lid * USTR + koff;
            const float* p0 = ptc + (size_t)(wave * 32 + lid) * CHUNK + koff;
            const float* p1 = ptc + (size_t)(wave * 32 + 16 + lid) * CHUNK + koff;
            v2f aU  = *(const v2f*)ur;
            v2f pb0 = *(const v2f*)p0;
            v2f pb1 = *(const v2f*)p1;
            #pragma unroll 2
            for (int j0 = 0; j0 < CHUNK; j0 += 4) {
                const int jn = (j0 + 4) & (CHUNK - 1);
                v2f aUn  = *(const v2f*)(ur + jn);
                v2f pb0n = *(const v2f*)(p0 + jn);
                v2f pb1n = *(const v2f*)(p1 + jn);
                y0 = wmma_f32(aU, pb0, y0);
                y1 = wmma_f32(aU, pb1, y1);
                aU = aUn; pb0 = pb0n; pb1 = pb1n;
            }
        }
        __syncthreads();                     // everyone done reading U
        #pragma unroll
        for (int v = 0; v < 8; ++v) {
            Ubuf[(v + 8 * half) * USTR + wave * 32 + lid]      = y0[v];
            Ubuf[(v + 8 * half) * USTR + wave * 32 + 16 + lid] = y1[v];
        }
        __syncthreads();

        // ---- out[b, c*64+i, row0+lid] = Y[r][i] + x  (transposed, coalesced) ----
        #pragma unroll
        for (int n = 0; n < 16; ++n) {
            const int i  = wave * 32 + 2 * n + half;
            const float yv = Ubuf[lid * USTR + i];
            const float xv = xc[(size_t)i * DDIM + row0 + lid];
            ob[(size_t)(c * CHUNK + i) * DDIM + row0 + lid] = yv + xv;
        }

        // ---- M = af*M - Z X, Z = Y diag(q); A-fragments (negated) preloaded ----
        v2f zf[16];
        #pragma unroll
        for (int n = 0; n < 16; ++n) {
            v2f u  = *(const v2f*)(Ubuf + lid * USTR + n * 4 + koff);
            v2f q2 = *(const v2f*)(qc + n * 4 + koff);
            zf[n].x = -(u.x * q2.x);
            zf[n].y = -(u.y * q2.y);
        }
        for (int t = 0; t < 16; ++t) {
            const int c0 = (wave * 16 + t) * 16;
            v8f acc;
            #pragma unroll
            for (int v = 0; v < 8; ++v)
                acc[v] = af * Mlds[(v + 8 * half) * DDIM + c0 + lid];
            v2f bx;
            bx.x = xc[(size_t)(koff)     * DDIM + c0 + lid];
            bx.y = xc[(size_t)(koff + 1) * DDIM + c0 + lid];
            #pragma unroll
            for (int n = 0; n < 16; ++n) {
                const int nn = (n + 1) & 15;
                v2f bxn;
                bxn.x = xc[(size_t)(nn * 4 + koff)     * DDIM + c0 + lid];
                bxn.y = xc[(size_t)(nn * 4 + koff + 1) * DDIM + c0 + lid];
                acc = wmma_f32(zf[n], bx, acc);
                bx = bxn;
            }
            #pragma unroll
            for (int v = 0; v < 8; ++v)
                Mlds[(v + 8 * half) * DDIM + c0 + lid] = acc[v];
        }
        __syncthreads();
    }
}

extern "C" void kernel_launch(void* const* d_in, const int* in_sizes, int n_in,
                              void* d_out, int out_size, void* d_ws, size_t ws_size,
                              hipStream_t stream) {
    (void)in_sizes; (void)n_in; (void)out_size; (void)ws_size;
    const float* x      = (const float*)d_in[0];
    const float* memory = (const float*)d_in[1];
    const float* a_w1   = (const float*)d_in[2];
    const float* a_b1   = (const float*)d_in[3];
    const float* a_w2   = (const float*)d_in[4];
    const float* a_b2   = (const float*)d_in[5];
    const float* e_w1   = (const float*)d_in[6];
    const float* e_b1   = (const float*)d_in[7];
    const float* e_w2   = (const float*)d_in[8];
    const float* e_b2   = (const float*)d_in[9];
    float* out = (float*)d_out;

    // Workspace layout (floats): alpha, eta, PT (per b,chunk 64x64), q, A_fin
    float* ws    = (float*)d_ws;
    float* alpha = ws;                                          // 65536
    float* eta   = alpha + (size_t)BATCH * SEQ;                 // 65536
    float* PTw   = eta   + (size_t)BATCH * SEQ;                 // 16*64*4096 = 4M
    float* qw    = PTw   + (size_t)BATCH * NCH * CHUNK * CHUNK; // 65536
    float* afw   = qw    + (size_t)BATCH * NCH * CHUNK;         // 1024

    gate_kernel<<<(BATCH * SEQ) / 16 / 8, 256, 0, stream>>>(
        x, a_w1, a_b1, a_w2, a_b2, e_w1, e_b1, e_w2, e_b2, alpha, eta);

    prep_kernel<<<BATCH * NCH, 256, 0, stream>>>(x, alpha, eta, PTw, qw, afw);

    scan_kernel<<<BATCH * 32, 64, 0, stream>>>(x, memory, PTw, qw, afw, out);
}